// ChunkedCEAutogradLoss_33148557590530
// MI455X (gfx1250) — compile-verified
//
#include <hip/hip_runtime.h>
#include <hip/hip_bf16.h>
#include <math.h>

// ---------------------------------------------------------------------------
// Fused linear + cross-entropy for MI455X (gfx1250, wave32).
//   logits = H @ W^T   (8192 x 32000, K=2048, fp32)
//   loss   = mean_t( logsumexp(logits[t]) - logits[t, target[t]] )
// Streaming logsumexp fused into an fp32 WMMA GEMM; logits never hit memory.
// v3: staged fragment loads (8 B + 16 A per unrolled step) issued before the
// 16 WMMAs that consume them -> clause-grouped loads + partial waits instead
// of full drains; __launch_bounds__(256,1) relaxes the VGPR budget.
// ---------------------------------------------------------------------------

typedef __attribute__((ext_vector_type(2))) float v2f;
typedef __attribute__((ext_vector_type(8))) float v8f;

#define DDIM        2048
#define VOCAB       32000
#define NTOK        8192              // 4 * 2048 tokens
#define MROWS       32                // tokens per workgroup (2 WMMA M-tiles)
#define MBLOCKS     (NTOK / MROWS)    // 256
#define NTILES      (VOCAB / 16)      // 2000 vocab tiles of 16
#define NCHUNK      10                // vocab chunks (26 MB of W each -> L2 resident)
#define TILES_PER_CHUNK (NTILES / NCHUNK)      // 200
#define TILES_PER_WAVE  (TILES_PER_CHUNK / 8)  // 25 (8 waves per workgroup)
#define LDS_STRIDE  2052              // 2048 + 4 pad -> conflict-free row access
#define KU          8                 // K-steps staged per unrolled iteration (K += 32)
#define IGNORE_IDX  (-100)

// ---------------------------------------------------------------------------
// Kernel 1: WMMA GEMM + per-(token, chunk) online logsumexp partials.
// grid = (MBLOCKS, NCHUNK), block = 256 (8 wave32).
// dynamic LDS = MROWS * LDS_STRIDE * 4 bytes (~257 KB of 320 KB).
// ---------------------------------------------------------------------------
__global__ __launch_bounds__(256, 1) void k_gemm_lse(const float* __restrict__ W,
                                                     const float* __restrict__ H,
                                                     float2* __restrict__ partial) {
    extern __shared__ float Alds[];                 // [MROWS][LDS_STRIDE]
    __shared__ float2 red[8][MROWS];                // per-wave row stats

    const int mblk  = blockIdx.x;                   // token block (fastest-varying)
    const int chunk = blockIdx.y;                   // vocab chunk
    const int tid   = threadIdx.x;

    // Cooperative load of A panel: 32 tokens x 2048 floats -> padded LDS.
    {
        const float4* src = (const float4*)(H + (size_t)mblk * MROWS * DDIM);
        for (int i = tid; i < MROWS * (DDIM / 4); i += 256) {
            const int row  = i / (DDIM / 4);
            const int col4 = i % (DDIM / 4);
            float4 v = src[i];
            *(float4*)&Alds[row * LDS_STRIDE + col4 * 4] = v;
        }
    }
    __syncthreads();

    const int wave = tid >> 5;
    const int lane = tid & 31;
    const int half = lane >> 4;          // 0: rows 0-7, 1: rows 8-15 (C layout)
    const int l16  = lane & 15;
    const int koff = half * 2;           // A/B fragment K-offset per half-wave

    float m0[8], s0[8], m1[8], s1[8];
#pragma unroll
    for (int r = 0; r < 8; ++r) {
        m0[r] = -INFINITY; s0[r] = 0.0f;
        m1[r] = -INFINITY; s1[r] = 0.0f;
    }

    const int tile0 = chunk * TILES_PER_CHUNK + wave * TILES_PER_WAVE;
    const float* arow0 = &Alds[l16 * LDS_STRIDE + koff];              // tokens 0..15
    const float* arow1 = arow0 + 16 * LDS_STRIDE;                     // tokens 16..31

    for (int t = 0; t < TILES_PER_WAVE; ++t) {
        const int nbase = (tile0 + t) * 16;
        const float* wrow = W + (size_t)(nbase + l16) * DDIM + koff;

        v8f c0 = {};                                 // fresh accumulators per tile
        v8f c1 = {};

#pragma unroll 1
        for (int k = 0; k < DDIM; k += 4 * KU) {
            // Stage ALL fragments for this step first: lets the scheduler
            // clause the loads and count waits down across the WMMAs.
            v2f bf[KU], a0f[KU], a1f[KU];
#pragma unroll
            for (int j = 0; j < KU; ++j)
                bf[j] = *(const v2f*)(wrow + k + 4 * j);    // global_load_b64 (L2-resident W)
#pragma unroll
            for (int j = 0; j < KU; ++j)
                a0f[j] = *(const v2f*)(arow0 + k + 4 * j);  // ds_load_b64
#pragma unroll
            for (int j = 0; j < KU; ++j)
                a1f[j] = *(const v2f*)(arow1 + k + 4 * j);  // ds_load_b64

            // Two WMMAs share each B fragment: 2x compute per L2 byte.
#pragma unroll
            for (int j = 0; j < KU; ++j) {
                c0 = __builtin_amdgcn_wmma_f32_16x16x4_f32(
                         false, a0f[j], false, bf[j], (short)0, c0, false, false);
                c1 = __builtin_amdgcn_wmma_f32_16x16x4_f32(
                         false, a1f[j], false, bf[j], (short)0, c1, false, false);
            }
        }

        // Online logsumexp update: lane holds column (nbase + l16) of 8 rows
        // in each of the two token tiles.
#pragma unroll
        for (int r = 0; r < 8; ++r) {
            float v, mo, mn;
            v  = c0[r];
            mo = m0[r];
            mn = fmaxf(mo, v);
            s0[r] = s0[r] * __expf(mo - mn) + __expf(v - mn);
            m0[r] = mn;

            v  = c1[r];
            mo = m1[r];
            mn = fmaxf(mo, v);
            s1[r] = s1[r] * __expf(mo - mn) + __expf(v - mn);
            m1[r] = mn;
        }
    }

    // Cross-lane merge within each 16-lane row group (wave32).
#pragma unroll
    for (int r = 0; r < 8; ++r) {
        float mm0 = m0[r], ss0 = s0[r];
        float mm1 = m1[r], ss1 = s1[r];
#pragma unroll
        for (int d = 1; d < 16; d <<= 1) {
            float mo, so, mn;
            mo = __shfl_xor(mm0, d, 16);
            so = __shfl_xor(ss0, d, 16);
            mn = fmaxf(mm0, mo);
            ss0 = ss0 * __expf(mm0 - mn) + so * __expf(mo - mn);
            mm0 = mn;

            mo = __shfl_xor(mm1, d, 16);
            so = __shfl_xor(ss1, d, 16);
            mn = fmaxf(mm1, mo);
            ss1 = ss1 * __expf(mm1 - mn) + so * __expf(mo - mn);
            mm1 = mn;
        }
        m0[r] = mm0; s0[r] = ss0;
        m1[r] = mm1; s1[r] = ss1;
    }

    // Lanes 0 and 16 hold reduced stats for rows half*8 + r of each tile.
    if (l16 == 0) {
#pragma unroll
        for (int r = 0; r < 8; ++r) {
            red[wave][half * 8 + r]      = make_float2(m0[r], s0[r]);
            red[wave][16 + half * 8 + r] = make_float2(m1[r], s1[r]);
        }
    }
    __syncthreads();

    // Merge the 8 waves' partials; one float2 per (token, chunk).
    if (tid < MROWS) {
        float mm = -INFINITY, ss = 0.0f;
#pragma unroll
        for (int w = 0; w < 8; ++w) {
            const float2 p = red[w][tid];
            const float mn = fmaxf(mm, p.x);
            ss = ss * __expf(mm - mn) + p.y * __expf(p.x - mn);
            mm = mn;
        }
        partial[(size_t)(mblk * MROWS + tid) * NCHUNK + chunk] = make_float2(mm, ss);
    }
}

// ---------------------------------------------------------------------------
// Kernel 2: target logit per token (one wave per token).
// ---------------------------------------------------------------------------
__global__ __launch_bounds__(256) void k_tgt(const float* __restrict__ W,
                                             const float* __restrict__ H,
                                             const int* __restrict__ T,
                                             float* __restrict__ tgtlogit) {
    const int wid  = (blockIdx.x * 256 + threadIdx.x) >> 5;
    const int lane = threadIdx.x & 31;
    if (wid >= NTOK) return;

    const int tgt = T[wid];
    const int ct  = min(max(tgt, 0), VOCAB - 1);
    const float4* w4 = (const float4*)(W + (size_t)ct * DDIM);
    const float4* h4 = (const float4*)(H + (size_t)wid * DDIM);

    float acc = 0.0f;
    for (int i = lane; i < DDIM / 4; i += 32) {
        const float4 a = h4[i];
        const float4 b = w4[i];
        acc += a.x * b.x + a.y * b.y + a.z * b.z + a.w * b.w;
    }
#pragma unroll
    for (int d = 16; d; d >>= 1) acc += __shfl_xor(acc, d, 32);
    if (lane == 0) tgtlogit[wid] = acc;
}

// ---------------------------------------------------------------------------
// Kernel 3: merge chunk partials -> LSE -> masked mean loss (single block).
// ---------------------------------------------------------------------------
__global__ __launch_bounds__(256) void k_final(const float2* __restrict__ partial,
                                               const float* __restrict__ tgtlogit,
                                               const int* __restrict__ T,
                                               float* __restrict__ out) {
    __shared__ float ssum[256];
    __shared__ float scnt[256];
    float total = 0.0f, cnt = 0.0f;

    for (int t = threadIdx.x; t < NTOK; t += 256) {
        float mm = -INFINITY, ss = 0.0f;
#pragma unroll
        for (int c = 0; c < NCHUNK; ++c) {
            const float2 p = partial[(size_t)t * NCHUNK + c];
            const float mn = fmaxf(mm, p.x);
            ss = ss * __expf(mm - mn) + p.y * __expf(p.x - mn);
            mm = mn;
        }
        const float lse = mm + __logf(ss);
        if (T[t] != IGNORE_IDX) { total += lse - tgtlogit[t]; cnt += 1.0f; }
    }
    ssum[threadIdx.x] = total;
    scnt[threadIdx.x] = cnt;
    __syncthreads();
    for (int d = 128; d; d >>= 1) {
        if (threadIdx.x < (unsigned)d) {
            ssum[threadIdx.x] += ssum[threadIdx.x + d];
            scnt[threadIdx.x] += scnt[threadIdx.x + d];
        }
        __syncthreads();
    }
    if (threadIdx.x == 0)
        out[0] = (scnt[0] > 0.0f) ? (ssum[0] / scnt[0]) : ssum[0];
}

// ---------------------------------------------------------------------------
extern "C" void kernel_launch(void* const* d_in, const int* in_sizes, int n_in,
                              void* d_out, int out_size, void* d_ws, size_t ws_size,
                              hipStream_t stream) {
    const float* W = (const float*)d_in[0];   // [32000, 2048] f32
    const float* H = (const float*)d_in[1];   // [4, 2048, 2048] f32
    const int*   T = (const int*)d_in[2];     // [4, 2048] int
    float* out = (float*)d_out;

    float2* partial  = (float2*)d_ws;                                    // [NTOK][NCHUNK]
    float*  tgtlogit = (float*)((char*)d_ws +
                                (size_t)NTOK * NCHUNK * sizeof(float2)); // [NTOK]

    const size_t lds_bytes = MROWS * LDS_STRIDE * sizeof(float);         // ~257 KB of 320 KB

    dim3 g1(MBLOCKS, NCHUNK);                    // mblk fastest -> W chunk stays in L2
    k_gemm_lse<<<g1, 256, lds_bytes, stream>>>(W, H, partial);

    k_tgt<<<NTOK / 8, 256, 0, stream>>>(W, H, T, tgtlogit);

    k_final<<<1, 256, 0, stream>>>(partial, tgtlogit, T, out);
}